// TrueHigherOrderAttention_27187142983714
// MI455X (gfx1250) — compile-verified
//
#include <hip/hip_runtime.h>
#include <hip/hip_bf16.h>

// Mask analysis: causal(j<=i, k<=j) ∧ window(j>=i, k>=i) forces j==k==i, so the
// softmax over (j,k) is an exact one-hot and the reference reduces to
//     out = ((x @ Wv1^T) ⊙ (x @ Wv2^T)) @ Wc^T
// Three fp32 GEMMs of 256x512x512 (~0.4 GFLOP, ~4MB traffic): latency-bound on
// MI455X, done with V_WMMA_F32_16X16X4_F32 to keep full fp32 fidelity.

typedef __attribute__((ext_vector_type(2))) float v2f;
typedef __attribute__((ext_vector_type(4))) float v4f;
typedef __attribute__((ext_vector_type(8))) float v8f;

namespace {

constexpr int TT   = 256;   // rows of x / output (B*T)
constexpr int CC   = 512;   // feature dim (K and N of every GEMM)
constexpr int KC   = 32;    // K chunk staged through LDS
constexpr int LDST = KC + 4; // LDS row stride (floats): 144B -> 16B aligned, banks spread

// C[M,N] = A[M,K] * W[N,K]^T   (optionally two W's with elementwise product of
// the two results, fused in the accumulator registers).
// Block: 256 threads = 8 waves. Block tile 32(M) x 64(N); wave tile 16x16.
template <bool DUAL>
__global__ __launch_bounds__(256) void gemm_xWT(const float* __restrict__ A,
                                                const float* __restrict__ W1,
                                                const float* __restrict__ W2,
                                                float* __restrict__ Out) {
  __shared__ float Als[32 * LDST];
  __shared__ float B1ls[64 * LDST];
  __shared__ float B2ls[DUAL ? 64 * LDST : 1];

  const int tid  = threadIdx.x;
  const int lane = tid & 31;
  const int wave = tid >> 5;      // 0..7
  const int wrow = wave >> 2;     // 0..1 -> which 16-row strip of the 32-row tile
  const int wcol = wave & 3;      // 0..3 -> which 16-col strip of the 64-col tile

  const int rowBase = blockIdx.y * 32;  // M
  const int colBase = blockIdx.x * 64;  // N

  const int mn    = lane & 15;    // A row (m) / B col (n) owned by this lane
  const int khalf = lane >> 4;    // selects K pair {0,1} vs {2,3} within each x4 step

  v8f acc1 = {0.f, 0.f, 0.f, 0.f, 0.f, 0.f, 0.f, 0.f};
  v8f acc2 = {0.f, 0.f, 0.f, 0.f, 0.f, 0.f, 0.f, 0.f};

  for (int kk = 0; kk < CC; kk += KC) {
    // ---- cooperative stage to LDS (all loads are aligned b128) ----
    {
      // A chunk: 32 x 32 floats = 256 float4, one per thread
      const int r  = tid >> 3;
      const int c4 = (tid & 7) << 2;
      v4f va = *(const v4f*)(A + (size_t)(rowBase + r) * CC + kk + c4);
      *(v4f*)(&Als[r * LDST + c4]) = va;
    }
#pragma unroll
    for (int rep = 0; rep < 2; ++rep) {
      // B chunk(s): 64 x 32 floats = 512 float4, two per thread
      const int idx = tid + rep * 256;
      const int r   = idx >> 3;
      const int c4  = (idx & 7) << 2;
      v4f vb1 = *(const v4f*)(W1 + (size_t)(colBase + r) * CC + kk + c4);
      *(v4f*)(&B1ls[r * LDST + c4]) = vb1;
      if (DUAL) {
        v4f vb2 = *(const v4f*)(W2 + (size_t)(colBase + r) * CC + kk + c4);
        *(v4f*)(&B2ls[r * LDST + c4]) = vb2;
      }
    }
    __syncthreads();

    // ---- 8 (or 16) WMMAs per chunk straight out of LDS ----
#pragma unroll
    for (int j = 0; j < KC; j += 4) {
      const int kc = j + 2 * khalf;
      v2f a  = *(const v2f*)(&Als[(wrow * 16 + mn) * LDST + kc]);
      v2f b1 = *(const v2f*)(&B1ls[(wcol * 16 + mn) * LDST + kc]);
      acc1 = __builtin_amdgcn_wmma_f32_16x16x4_f32(false, a, false, b1,
                                                   (short)0, acc1, false, false);
      if (DUAL) {
        v2f b2 = *(const v2f*)(&B2ls[(wcol * 16 + mn) * LDST + kc]);
        acc2 = __builtin_amdgcn_wmma_f32_16x16x4_f32(false, a, false, b2,
                                                     (short)0, acc2, false, false);
      }
    }
    __syncthreads();
  }

  // ---- epilogue: C/D layout is (m = v + 8*khalf, n = lane&15) in VGPR v ----
  float* outp = Out + (size_t)(rowBase + wrow * 16 + 8 * khalf) * CC +
                colBase + wcol * 16 + mn;
#pragma unroll
  for (int v = 0; v < 8; ++v) {
    float val = acc1[v];
    if (DUAL) val *= acc2[v];
    outp[(size_t)v * CC] = val;
  }
}

}  // namespace

extern "C" void kernel_launch(void* const* d_in, const int* in_sizes, int n_in,
                              void* d_out, int out_size, void* d_ws, size_t ws_size,
                              hipStream_t stream) {
  (void)in_sizes; (void)n_in; (void)out_size; (void)ws_size;
  // setup_inputs order: x, Wq, Wk1, Wk2, Wv1, Wv2, Wc  (q/k projections are dead)
  const float* x   = (const float*)d_in[0];
  const float* Wv1 = (const float*)d_in[4];
  const float* Wv2 = (const float*)d_in[5];
  const float* Wc  = (const float*)d_in[6];
  float* out = (float*)d_out;
  float* P   = (float*)d_ws;  // 256*512 fp32 = 512 KB scratch

  dim3 block(256);
  dim3 grid(CC / 64, TT / 32);  // (8, 8)

  // P = (x @ Wv1^T) ⊙ (x @ Wv2^T)
  gemm_xWT<true><<<grid, block, 0, stream>>>(x, Wv1, Wv2, P);
  // out = P @ Wc^T
  gemm_xWT<false><<<grid, block, 0, stream>>>(P, Wc, nullptr, out);
}